// fab_penalty_ls_gap_39178691674436
// MI455X (gfx1250) — compile-verified
//
#include <hip/hip_runtime.h>
#include <math.h>

#define N 6144
#define TILE_X 64
#define TILE_Y 4
#define ROWS_PER_THREAD 8
#define TILE_ROWS (TILE_Y * ROWS_PER_THREAD)   // 32 rows per block
#define NBX (N / TILE_X)                       // 96
#define NBY (N / TILE_ROWS)                    // 192
#define NBLOCKS (NBX * NBY)                    // 18432 partials (72 KB in d_ws)

typedef __attribute__((ext_vector_type(2))) float v2f;
typedef __attribute__((ext_vector_type(8))) float v8f;

// ---- first-derivative helpers (numpy.gradient, edge_order=1) ----
template<bool EDGE>
__device__ __forceinline__ float gx(const float* __restrict__ p, int i, int j, float invd) {
  if (EDGE) {
    if (i == 0)     return (p[N + j]         - p[j]) * invd;
    if (i == N - 1) return (p[(N-1)*N + j]   - p[(N-2)*N + j]) * invd;
  }
  return (p[(i+1)*N + j] - p[(i-1)*N + j]) * (0.5f * invd);
}

template<bool EDGE>
__device__ __forceinline__ float gy(const float* __restrict__ p, int i, int j, float invd) {
  if (EDGE) {
    if (j == 0)     return (p[i*N + 1]       - p[i*N]) * invd;
    if (j == N - 1) return (p[i*N + (N-1)]   - p[i*N + (N-2)]) * invd;
  }
  return (p[i*N + j + 1] - p[i*N + j - 1]) * (0.5f * invd);
}

// ---- full penalty at one point ----
// Note: the reference adds SC=1e-12 to phi_x/phi_y BEFORE the second gradient,
// but (a+SC)-(b+SC) == a-b to fp32 rounding for these O(1) gradients, so the
// SC terms are dropped inside the differences and kept only in px/py.
template<bool EDGE>
__device__ __forceinline__ float point(const float* __restrict__ p, int i, int j,
                                       float invd, float d2, float pi_d) {
  const float SC = 1e-12f;

  float px = gx<EDGE>(p, i, j, invd) + SC;
  float py = gy<EDGE>(p, i, j, invd) + SC;

  // phi_xx: gradient along axis0 of phi_x
  float pxx;
  if (EDGE && i == 0)
    pxx = (gx<EDGE>(p, 1, j, invd)   - gx<EDGE>(p, 0, j, invd))   * invd;
  else if (EDGE && i == N - 1)
    pxx = (gx<EDGE>(p, N-1, j, invd) - gx<EDGE>(p, N-2, j, invd)) * invd;
  else
    pxx = (gx<EDGE>(p, i+1, j, invd) - gx<EDGE>(p, i-1, j, invd)) * (0.5f * invd);

  // phi_xy: gradient along axis1 of phi_x
  float pxy;
  if (EDGE && j == 0)
    pxy = (gx<EDGE>(p, i, 1, invd)   - gx<EDGE>(p, i, 0, invd))   * invd;
  else if (EDGE && j == N - 1)
    pxy = (gx<EDGE>(p, i, N-1, invd) - gx<EDGE>(p, i, N-2, invd)) * invd;
  else
    pxy = (gx<EDGE>(p, i, j+1, invd) - gx<EDGE>(p, i, j-1, invd)) * (0.5f * invd);

  // phi_yy: gradient along axis1 of phi_y
  float pyy;
  if (EDGE && j == 0)
    pyy = (gy<EDGE>(p, i, 1, invd)   - gy<EDGE>(p, i, 0, invd))   * invd;
  else if (EDGE && j == N - 1)
    pyy = (gy<EDGE>(p, i, N-1, invd) - gy<EDGE>(p, i, N-2, invd)) * invd;
  else
    pyy = (gy<EDGE>(p, i, j+1, invd) - gy<EDGE>(p, i, j-1, invd)) * (0.5f * invd);

  float phiv = p[i*N + j];
  float px2  = px * px;
  float py2  = py * py;
  float s    = px2 + py2;
  float v    = fmaxf(__builtin_amdgcn_sqrtf(s), 1e-8f);
  float num  = fmaf(px2, pxx, fmaf(py2, pyy, 2.0f * px * py * pxy));

  // penalty arg = |num| / (v^2 * (pi_d*|phi| + v));  single rcp + 1 NR step
  float den  = (v * v) * fmaf(pi_d, fabsf(phiv), v);
  float r    = __builtin_amdgcn_rcpf(den);
  r          = r * fmaf(-den, r, 2.0f);              // Newton: ~0.5 ulp
  float pen  = fmaxf(fmaf(fabsf(num), r, -pi_d), 0.0f) * d2;
  return (pen == pen) ? pen : 0.0f;                  // nansum: NaN contributes 0
}

__global__ __launch_bounds__(256)
void fab_penalty_kernel(const float* __restrict__ phi, const int* __restrict__ gsz,
                        float* __restrict__ partials) {
  const float d    = (float)(*gsz);
  const float invd = 1.0f / d;
  const float d2   = d * d;
  const float pi_d = (float)(3.14159265358979323846 / 1.3);   // BETA=1, MIN_FEATURE_SIZE=1

  const int j  = blockIdx.x * TILE_X + threadIdx.x;
  const int i0 = blockIdx.y * TILE_ROWS + threadIdx.y * ROWS_PER_THREAD;

  // Streaming hint for the first (compulsory-HBM) pass; later halo reuse hits L2.
  {
    int ipf = i0 + ROWS_PER_THREAD + 2;
    ipf = (ipf < N) ? ipf : (N - 1);
    __builtin_prefetch(&phi[ipf * N + j], 0, 1);     // -> global_prefetch_b8
  }

  const bool edge = (blockIdx.x == 0) || (blockIdx.x == NBX - 1) ||
                    (blockIdx.y == 0) || (blockIdx.y == NBY - 1);

  float acc = 0.0f;
  if (!edge) {
#pragma unroll
    for (int r = 0; r < ROWS_PER_THREAD; ++r)
      acc += point<false>(phi, i0 + r, j, invd, d2, pi_d);
  } else {
#pragma unroll 1   // branch-heavy path, executed by ~3% of blocks: keep it small
    for (int r = 0; r < ROWS_PER_THREAD; ++r)
      acc += point<true>(phi, i0 + r, j, invd, d2, pi_d);
  }

  // ---- wave32 reduction via V_WMMA_F32_16X16X4_F32 with B = ones ----
  // A[m,k]: lane m (0..15) VGPR0 -> K=0, lanes 16..31 VGPR0 -> K=2 (VGPR1 zeroed).
  // D[m,n] = sum_k A[m,k] = acc(m) + acc(m+16) for every column n.
  // Each lane sums its 8 D-VGPRs -> lane<16 holds sum over M=0..7, lane>=16 over M=8..15.
  // Wave total = that value at lane 0 + at lane 16.
  float wave_sum;
#if __has_builtin(__builtin_amdgcn_wmma_f32_16x16x4_f32)
  {
    v2f a; a[0] = acc;  a[1] = 0.0f;
    v2f b; b[0] = 1.0f; b[1] = 1.0f;
    v8f c = {};
    c = __builtin_amdgcn_wmma_f32_16x16x4_f32(false, a, false, b, (short)0, c, false, false);
    float s = c[0] + c[1] + c[2] + c[3] + c[4] + c[5] + c[6] + c[7];
    wave_sum = __shfl(s, 0, 32) + __shfl(s, 16, 32);
  }
#else
  {
    float s = acc;
    for (int off = 16; off > 0; off >>= 1) s += __shfl_down(s, off, 32);
    wave_sum = __shfl(s, 0, 32);
  }
#endif

  __shared__ float wsums[256 / 32];
  const int flat = threadIdx.y * TILE_X + threadIdx.x;
  const int wid  = flat >> 5;
  const int lane = flat & 31;
  if (lane == 0) wsums[wid] = wave_sum;
  __syncthreads();
  if (flat == 0) {
    float t = 0.0f;
#pragma unroll
    for (int w = 0; w < 256 / 32; ++w) t += wsums[w];
    partials[blockIdx.y * NBX + blockIdx.x] = t;
  }
}

// Deterministic fixed-order final reduction of the 18432 block partials.
__global__ __launch_bounds__(256)
void fab_reduce_kernel(const float* __restrict__ partials, float* __restrict__ out) {
  __shared__ float sm[256];
  float t = 0.0f;
  for (int idx = threadIdx.x; idx < NBLOCKS; idx += 256) t += partials[idx];
  sm[threadIdx.x] = t;
  __syncthreads();
  for (int s = 128; s > 0; s >>= 1) {
    if (threadIdx.x < s) sm[threadIdx.x] += sm[threadIdx.x + s];
    __syncthreads();
  }
  if (threadIdx.x == 0) out[0] = sm[0];
}

extern "C" void kernel_launch(void* const* d_in, const int* in_sizes, int n_in,
                              void* d_out, int out_size, void* d_ws, size_t ws_size,
                              hipStream_t stream) {
  const float* phi = (const float*)d_in[0];
  const int*   gsz = (const int*)d_in[1];
  float* partials  = (float*)d_ws;          // NBLOCKS floats = 72 KB
  float* out       = (float*)d_out;

  dim3 block(TILE_X, TILE_Y);
  dim3 grid(NBX, NBY);
  fab_penalty_kernel<<<grid, block, 0, stream>>>(phi, gsz, partials);
  fab_reduce_kernel<<<1, 256, 0, stream>>>(partials, out);
}